// adder2d2_0_36910948942379
// MI455X (gfx1250) — compile-verified
//
#include <hip/hip_runtime.h>
#include <stdint.h>

#define IN_CH  64
#define OUT_CH 64
#define HW     32

#if defined(__gfx1250__)
#  if __has_builtin(__builtin_amdgcn_tensor_load_to_lds)
#    define USE_TDM 1
#  endif
#  if __has_builtin(__builtin_amdgcn_s_wait_tensorcnt)
#    define HAVE_WAIT_TENSOR_BUILTIN 1
#  endif
#endif

#if USE_TDM
typedef unsigned int v4u __attribute__((ext_vector_type(4)));
typedef int          v8i __attribute__((ext_vector_type(8)));
typedef int          v4i __attribute__((ext_vector_type(4)));
#endif

// Block: 256 threads (8 wave32). blockIdx.x = n*16 + row_tile (2 output rows
// per tile), blockIdx.y = output-channel half (32 channels each).
// LDS input tile: [64 ch][4 rows][34 cols] fp32 with halo baked in.
__global__ __launch_bounds__(256) void adder2d_tdm_kernel(
    const float* __restrict__ x,   // (4, 64, 32, 32)
    const float* __restrict__ w,   // (64, 64, 3, 3)
    float* __restrict__ out) {     // (4, 64, 32, 32)
  __shared__ __align__(16) float sx[IN_CH * 4 * 34];

  const int tid   = threadIdx.x;
  const int n     = blockIdx.x >> 4;
  const int rt    = blockIdx.x & 15;
  const int r0    = rt * 2;          // first output row of this tile
  const int ohalf = blockIdx.y;

  // Zero LDS: provides the zero halo (cols 0/33 and missing boundary rows).
  for (int i = tid; i < IN_CH * 4 * 34; i += 256) sx[i] = 0.0f;
  __syncthreads();

  const int gr0   = (r0 == 0) ? 0 : (r0 - 1);          // first valid input row
  const int grend = (r0 + 2 > HW - 1) ? (HW - 1) : (r0 + 2);
  const int nrows = grend - gr0 + 1;                   // 3 at edges, else 4
  const int lrow0 = (r0 == 0) ? 1 : 0;                 // local row of gr0

#if USE_TDM
  if (tid < 32) {  // single wave issues the TDM descriptor (EXEC ignored)
    const unsigned lds_off =
        (unsigned)(unsigned long long)(uintptr_t)(&sx[lrow0 * 34 + 1]);
    const unsigned long long ga = (unsigned long long)(uintptr_t)(
        x + ((size_t)n * IN_CH * HW * HW + (size_t)gr0 * HW));

    // D# group 0: count=1, lds_addr, global_addr, type=2
    v4u g0;
    g0.x = 1u;
    g0.y = lds_off;
    g0.z = (unsigned)(ga & 0xFFFFFFFFull);
    g0.w = (unsigned)((ga >> 32) & 0x1FFFFFFull) | (2u << 30);

    // D# group 1: data_size=4B, iterate_enable, pad_enable,
    // pad_interval=32 dwords (enc 4), pad_amount=2 dwords (enc 1),
    // tensor_dim0=32, tensor_dim1=nrows, tile_dim0=32, tile_dim1=nrows,
    // tensor_dim0_stride=32 (row pitch in elements)
    v8i g1;
    g1[0] = (int)((2u << 16) | (1u << 19) | (1u << 20) | (4u << 22) | (1u << 25));
    g1[1] = (int)(32u << 16);                 // tensor_dim0[15:0] (bits 63:48)
    g1[2] = (int)(((unsigned)nrows) << 16);   // tensor_dim1[15:0]
    g1[3] = (int)(32u << 16);                 // tile_dim0
    g1[4] = nrows;                            // tile_dim1
    g1[5] = HW;                               // tensor_dim0_stride lo32
    g1[6] = 0;
    g1[7] = 0;

    // D# group 2 (iterate mode): lds_addr_increment = 4*34 elements,
    // global_addr_increment = 32*32 elements (channel stride),
    // iterate_count = 63 -> 64 iterations (one per input channel)
    v4i g2;
    g2[0] = 0;
    g2[1] = 4 * 34;
    g2[2] = HW * HW;
    g2[3] = (63 << 16);
    v4i g3 = {0, 0, 0, 0};

    // 6-arg form (amdgpu-toolchain clang-23):
    // (uint32x4 g0, int32x8 g1, int32x4 g2, int32x4 g3, int32x8 extra, i32 cpol)
    v8i g4 = {0, 0, 0, 0, 0, 0, 0, 0};
    __builtin_amdgcn_tensor_load_to_lds(g0, g1, g2, g3, g4, 0);
#if HAVE_WAIT_TENSOR_BUILTIN
    __builtin_amdgcn_s_wait_tensorcnt(0);
#else
    asm volatile("s_wait_tensorcnt 0" ::: "memory");
#endif
  }
#else
  // Fallback staging path (host pass / toolchains without the TDM builtin)
  for (int q = tid; q < IN_CH * nrows * HW; q += 256) {
    const int c   = q / (nrows * HW);
    const int rem = q - c * (nrows * HW);
    const int i   = rem / HW;
    const int col = rem - i * HW;
    sx[(c * 4 + lrow0 + i) * 34 + 1 + col] =
        x[((size_t)(n * IN_CH + c) * HW + (gr0 + i)) * HW + col];
  }
#endif
  __syncthreads();

  // Thread mapping: lane -> output channel (within half), tid>>5 is
  // wave-uniform -> all lanes of a wave broadcast-read the same LDS words.
  const int o  = (tid & 31) + (ohalf << 5);
  const int p  = tid >> 5;   // 0..7, uniform per wave
  const int r  = p >> 2;     // output row within the 2-row tile
  const int cg = p & 3;      // 8-column group

  const float* __restrict__ wrow = w + (size_t)o * (IN_CH * 9);

  float acc[8];
#pragma unroll
  for (int j = 0; j < 8; ++j) acc[j] = 0.0f;

  for (int c = 0; c < IN_CH; ++c) {
    float wv[9];
#pragma unroll
    for (int t = 0; t < 9; ++t) wv[t] = wrow[c * 9 + t];

    // 3 LDS rows x 10 cols cover all 9 taps for this thread's 8 pixels.
    float xr[3][10];
#pragma unroll
    for (int t = 0; t < 3; ++t) {
      const float* s = &sx[(c * 4 + r + t) * 34 + cg * 8];
#pragma unroll
      for (int j = 0; j < 10; ++j) xr[t][j] = s[j];
    }

#pragma unroll
    for (int kh = 0; kh < 3; ++kh) {
#pragma unroll
      for (int kw = 0; kw < 3; ++kw) {
        const float wvv = wv[kh * 3 + kw];
#pragma unroll
        for (int j = 0; j < 8; ++j)
          acc[j] += __builtin_fabsf(wvv - xr[kh][j + kw]);
      }
    }
  }

  float* outp =
      out + (((size_t)n * OUT_CH + o) * HW + (r0 + r)) * HW + cg * 8;
  float4 lo = make_float4(-acc[0], -acc[1], -acc[2], -acc[3]);
  float4 hi = make_float4(-acc[4], -acc[5], -acc[6], -acc[7]);
  *(float4*)(outp)     = lo;
  *(float4*)(outp + 4) = hi;
}

extern "C" void kernel_launch(void* const* d_in, const int* in_sizes, int n_in,
                              void* d_out, int out_size, void* d_ws, size_t ws_size,
                              hipStream_t stream) {
  (void)in_sizes; (void)n_in; (void)out_size; (void)d_ws; (void)ws_size;
  const float* x = (const float*)d_in[0];   // (4, 64, 32, 32) fp32
  const float* w = (const float*)d_in[1];   // (64, 64, 3, 3) fp32
  float* out     = (float*)d_out;           // (4, 64, 32, 32) fp32

  dim3 grid(64, 2, 1);   // 4 images * 16 row-tiles, 2 channel halves
  adder2d_tdm_kernel<<<grid, 256, 0, stream>>>(x, w, out);
}